// GCN_7937099563688
// MI455X (gfx1250) — compile-verified
//
#include <hip/hip_runtime.h>
#include <cstdint>
#include <cstddef>

typedef __attribute__((ext_vector_type(2))) float v2f;
typedef __attribute__((ext_vector_type(8))) float v8f;

#define NFEAT 100
#define HID   64
#define NCLS  47
#define NCLSP 48

// ---------------- degree / norm ----------------

__global__ void __launch_bounds__(256) k_set_deg(float* deg, int n) {
  int i = blockIdx.x * 256 + threadIdx.x;
  if (i < n) deg[i] = 1.0f;  // self loop
}

__global__ void __launch_bounds__(256) k_add_deg(const long long* __restrict__ col,
                                                 float* deg, long long E) {
  long long e = (long long)blockIdx.x * 256 + threadIdx.x;
  if (e < E) atomicAdd(&deg[(int)col[e]], 1.0f);
}

__global__ void __launch_bounds__(256) k_rsqrt(float* d, int n) {
  int i = blockIdx.x * 256 + threadIdx.x;
  if (i < n) d[i] = rsqrtf(d[i]);
}

// ---------------- GEMM 1: XW = X[N,100] @ W0[100,64] (fp32 WMMA) ----------------
// One wave per 16-row tile; 4 accumulator tiles cover N=64; K stepped by 4.

__global__ void __launch_bounds__(256) k_gemm1(const float* __restrict__ X,
                                               const float* __restrict__ W,
                                               float* __restrict__ XW, int mtiles) {
  int wave = threadIdx.x >> 5;
  int tile = blockIdx.x * 8 + wave;
  if (tile >= mtiles) return;           // wave-uniform exit: EXEC stays all-1s for WMMA
  int lane = threadIdx.x & 31;
  int m    = lane & 15;
  int kh   = (lane >> 4) << 1;          // 0 or 2: K-half within the 4-wide step
  int row0 = tile << 4;

  const float* arow = X + (size_t)(row0 + m) * NFEAT + kh;
  v8f c0 = {}, c1 = {}, c2 = {}, c3 = {};

  for (int k = 0; k < NFEAT; k += 4) {
    float2 av = *reinterpret_cast<const float2*>(arow + k);   // 8B-aligned (k+kh even)
    v2f a; a.x = av.x; a.y = av.y;
    const float* bp = W + (size_t)(k + kh) * HID + m;         // rows k+kh, k+kh+1
    v2f b;
    b.x = bp[0];        b.y = bp[HID];
    c0 = __builtin_amdgcn_wmma_f32_16x16x4_f32(false, a, false, b, (short)0, c0, false, false);
    b.x = bp[16];       b.y = bp[HID + 16];
    c1 = __builtin_amdgcn_wmma_f32_16x16x4_f32(false, a, false, b, (short)0, c1, false, false);
    b.x = bp[32];       b.y = bp[HID + 32];
    c2 = __builtin_amdgcn_wmma_f32_16x16x4_f32(false, a, false, b, (short)0, c2, false, false);
    b.x = bp[48];       b.y = bp[HID + 48];
    c3 = __builtin_amdgcn_wmma_f32_16x16x4_f32(false, a, false, b, (short)0, c3, false, false);
  }

  int half = (lane >> 4) << 3;          // C/D: vgpr r holds row r + 8*half
  float* crow = XW + (size_t)row0 * HID + m;
#pragma unroll
  for (int r = 0; r < 8; ++r) {
    float* cp = crow + (size_t)(r + half) * HID;
    cp[0]  = c0[r];
    cp[16] = c1[r];
    cp[32] = c2[r];
    cp[48] = c3[r];
  }
}

// ---------------- GEMM 2: HW = relu(ACC + b0) @ W1[64,47], output stride 48 ----------------

__global__ void __launch_bounds__(256) k_gemm2(const float* __restrict__ Hacc,
                                               const float* __restrict__ b0,
                                               const float* __restrict__ W,
                                               float* __restrict__ HW, int mtiles) {
  int wave = threadIdx.x >> 5;
  int tile = blockIdx.x * 8 + wave;
  if (tile >= mtiles) return;
  int lane = threadIdx.x & 31;
  int m    = lane & 15;
  int kh   = (lane >> 4) << 1;
  int row0 = tile << 4;

  const float* arow = Hacc + (size_t)(row0 + m) * HID + kh;
  v8f c0 = {}, c1 = {}, c2 = {};

  for (int k = 0; k < HID; k += 4) {
    float2 av = *reinterpret_cast<const float2*>(arow + k);
    v2f a;
    a.x = fmaxf(av.x + b0[k + kh],     0.0f);   // fused bias + ReLU on A operand
    a.y = fmaxf(av.y + b0[k + kh + 1], 0.0f);
    const float* bp = W + (size_t)(k + kh) * NCLS;
    const float* bq = bp + NCLS;
    v2f b;
    b.x = bp[m];              b.y = bq[m];
    c0 = __builtin_amdgcn_wmma_f32_16x16x4_f32(false, a, false, b, (short)0, c0, false, false);
    b.x = bp[m + 16];         b.y = bq[m + 16];
    c1 = __builtin_amdgcn_wmma_f32_16x16x4_f32(false, a, false, b, (short)0, c1, false, false);
    int n2 = m + 32;                              // 32..47; col 47 is padding -> 0
    b.x = (n2 < NCLS) ? bp[n2] : 0.0f;
    b.y = (n2 < NCLS) ? bq[n2] : 0.0f;
    c2 = __builtin_amdgcn_wmma_f32_16x16x4_f32(false, a, false, b, (short)0, c2, false, false);
  }

  int half = (lane >> 4) << 3;
  float* crow = HW + (size_t)row0 * NCLSP + m;
#pragma unroll
  for (int r = 0; r < 8; ++r) {
    float* cp = crow + (size_t)(r + half) * NCLSP;
    cp[0]  = c0[r];
    cp[16] = c1[r];
    cp[32] = c2[r];                                // col 47 stores 0 (padding)
  }
}

// ---------------- aggregation (self-loop init + edge scatter) ----------------

__global__ void __launch_bounds__(256) k_self1(const float* __restrict__ dinv,
                                               const float* __restrict__ xw,
                                               float* __restrict__ acc, int n) {
  long long idx = (long long)blockIdx.x * 256 + threadIdx.x;
  if (idx >= (long long)n * HID) return;
  int i = (int)(idx >> 6);
  float di = dinv[i];
  acc[idx] = di * di * xw[idx];
}

__global__ void __launch_bounds__(256) k_scat1(const long long* __restrict__ row,
                                               const long long* __restrict__ col,
                                               const float* __restrict__ dinv,
                                               const float* __restrict__ xw,
                                               float* acc, long long E) {
  long long idx = (long long)blockIdx.x * 256 + threadIdx.x;
  if (idx >= E * HID) return;
  long long e = idx >> 6;
  int f = (int)(idx & (HID - 1));
  int r = (int)row[e], c = (int)col[e];
  float nrm = dinv[r] * dinv[c];
  atomicAdd(&acc[(size_t)c * HID + f], nrm * xw[(size_t)r * HID + f]);
}

__global__ void __launch_bounds__(256) k_self2(const float* __restrict__ dinv,
                                               const float* __restrict__ hw,
                                               const float* __restrict__ b1,
                                               float* __restrict__ out, int n) {
  long long idx = (long long)blockIdx.x * 256 + threadIdx.x;
  if (idx >= (long long)n * NCLS) return;
  int i = (int)(idx / NCLS);
  int f = (int)(idx % NCLS);
  float di = dinv[i];
  out[idx] = di * di * hw[(size_t)i * NCLSP + f] + b1[f];
}

__global__ void __launch_bounds__(256) k_scat2(const long long* __restrict__ row,
                                               const long long* __restrict__ col,
                                               const float* __restrict__ dinv,
                                               const float* __restrict__ hw,
                                               float* out, long long E) {
  long long idx = (long long)blockIdx.x * 256 + threadIdx.x;
  if (idx >= E * NCLSP) return;
  long long e = idx / NCLSP;
  int f = (int)(idx % NCLSP);
  if (f >= NCLS) return;
  int r = (int)row[e], c = (int)col[e];
  float nrm = dinv[r] * dinv[c];
  atomicAdd(&out[(size_t)c * NCLS + f], nrm * hw[(size_t)r * NCLSP + f]);
}

// ---------------- launch ----------------

extern "C" void kernel_launch(void* const* d_in, const int* in_sizes, int n_in,
                              void* d_out, int out_size, void* d_ws, size_t ws_size,
                              hipStream_t stream) {
  const float*     x   = (const float*)d_in[0];
  const long long* ei  = (const long long*)d_in[1];   // [2, E] int64
  const float*     W0  = (const float*)d_in[2];
  const float*     b0  = (const float*)d_in[3];
  const float*     W1  = (const float*)d_in[4];
  const float*     b1  = (const float*)d_in[5];
  float*           out = (float*)d_out;

  int       N = in_sizes[0] / NFEAT;
  long long E = (long long)in_sizes[1] / 2;
  const long long* rowp = ei;       // edge_index[0] = source
  const long long* colp = ei + E;   // edge_index[1] = dest

  // workspace: dinv[N] | xw[N*HID] (reused as hw, stride NCLSP) | acc[N*HID]
  float* dinv = (float*)d_ws;
  float* xw   = dinv + N;
  float* acc  = xw + (size_t)N * HID;
  float* hw   = xw;                 // alias: xw dead after scatter1

  int mtiles = (N + 15) / 16;
  dim3 blk(256);

  k_set_deg<<<(N + 255) / 256, blk, 0, stream>>>(dinv, N);
  k_add_deg<<<(unsigned)((E + 255) / 256), blk, 0, stream>>>(colp, dinv, E);
  k_rsqrt<<<(N + 255) / 256, blk, 0, stream>>>(dinv, N);

  k_gemm1<<<(mtiles + 7) / 8, blk, 0, stream>>>(x, W0, xw, mtiles);

  long long t1 = (long long)N * HID;
  k_self1<<<(unsigned)((t1 + 255) / 256), blk, 0, stream>>>(dinv, xw, acc, N);
  long long t2 = E * HID;
  k_scat1<<<(unsigned)((t2 + 255) / 256), blk, 0, stream>>>(rowp, colp, dinv, xw, acc, E);

  k_gemm2<<<(mtiles + 7) / 8, blk, 0, stream>>>(acc, b0, W1, hw, mtiles);

  long long t3 = (long long)N * NCLS;
  k_self2<<<(unsigned)((t3 + 255) / 256), blk, 0, stream>>>(dinv, hw, b1, out, N);
  long long t4 = E * NCLSP;
  k_scat2<<<(unsigned)((t4 + 255) / 256), blk, 0, stream>>>(rowp, colp, dinv, hw, out, E);
}